// TreeLikelihood_88605175317054
// MI455X (gfx1250) — compile-verified
//
#include <hip/hip_runtime.h>
#include <math.h>

typedef __attribute__((ext_vector_type(2))) float        v2f;
typedef __attribute__((ext_vector_type(8))) float        v8f;
typedef __attribute__((ext_vector_type(4))) unsigned int v4u;
typedef __attribute__((ext_vector_type(4))) int          v4i;
typedef __attribute__((ext_vector_type(8))) int          v8i;

#define NUM_LEAVES 2048
#define NUM_SITES  256
#define NUM_STATES 20
#define NUM_NODES  4095
#define NUM_OPS    2047
// Swizzled P per node: 2 M-tiles * 5 K-chunks * 32 lanes * 2 floats
#define PSWZ_PER_NODE 640
#define PSWZ_TOTAL ((size_t)NUM_NODES * PSWZ_PER_NODE)
#define TILE_FLOATS 320   // 16 sites * 20 states staged per step

// ---------------------------------------------------------------------------
// Kernel A: per-branch P = expm(Q * t), stored pre-swizzled for WMMA A-operand.
// ---------------------------------------------------------------------------
__global__ __launch_bounds__(256) void build_p_kernel(
    const float* __restrict__ exch, const float* __restrict__ stat_logits,
    const float* __restrict__ blen, float* __restrict__ pswz)
{
    __shared__ float Qs[400], Ts[400], Ws[400], piS[20];
    __shared__ float scaleS;
    const int tid = threadIdx.x;
    const int node = blockIdx.x;

    if (tid == 0) {
        float mx = -1e30f;
        for (int i = 0; i < 20; ++i) mx = fmaxf(mx, stat_logits[i]);
        float sum = 0.f;
        for (int i = 0; i < 20; ++i) { float e = __expf(stat_logits[i] - mx); piS[i] = e; sum += e; }
        float inv = 1.f / sum;
        for (int i = 0; i < 20; ++i) piS[i] *= inv;
    }
    __syncthreads();

    for (int e = tid; e < 400; e += 256) {
        int i = e / 20, j = e % 20;
        float s = (i == j) ? 0.f : 0.5f * (exch[i * 20 + j] + exch[j * 20 + i]);
        Qs[e] = s * piS[j];
    }
    __syncthreads();
    if (tid < 20) {
        float rs = 0.f;
        for (int j = 0; j < 20; ++j) if (j != tid) rs += Qs[tid * 20 + j];
        Qs[tid * 21] = -rs;
    }
    __syncthreads();
    if (tid == 0) {
        float rate = 0.f;
        for (int i = 0; i < 20; ++i) rate -= piS[i] * Qs[i * 21];
        scaleS = blen[node] / (rate * 32.0f);
    }
    __syncthreads();
    for (int e = tid; e < 400; e += 256) Qs[e] *= scaleS;
    for (int e = tid; e < 400; e += 256) Ts[e] = (e / 20 == e % 20) ? 1.f : 0.f;
    __syncthreads();

    for (int m = 10; m >= 1; --m) {
        for (int e = tid; e < 400; e += 256) {
            int i = e / 20, j = e % 20;
            float acc = 0.f;
            for (int k = 0; k < 20; ++k) acc += Qs[i * 20 + k] * Ts[k * 20 + j];
            Ws[e] = acc;
        }
        __syncthreads();
        float invm = 1.f / (float)m;
        for (int e = tid; e < 400; e += 256)
            Ts[e] = ((e / 20 == e % 20) ? 1.f : 0.f) + Ws[e] * invm;
        __syncthreads();
    }
    for (int s = 0; s < 5; ++s) {
        for (int e = tid; e < 400; e += 256) {
            int i = e / 20, j = e % 20;
            float acc = 0.f;
            for (int k = 0; k < 20; ++k) acc += Ts[i * 20 + k] * Ts[k * 20 + j];
            Ws[e] = acc;
        }
        __syncthreads();
        for (int e = tid; e < 400; e += 256) Ts[e] = Ws[e];
        __syncthreads();
    }

    float* dst = pswz + (size_t)node * PSWZ_PER_NODE;
    for (int e = tid; e < PSWZ_PER_NODE; e += 256) {
        int q    = e & 1;
        int lane = (e >> 1) & 31;
        int c    = (e >> 6) % 5;
        int mt   = e / 320;
        int row  = mt * 16 + (lane & 15);
        int col  = 4 * c + ((lane < 16) ? 0 : 2) + q;
        dst[e] = (row < 20) ? Ts[row * 20 + col] : 0.f;
    }
}

// ---------------------------------------------------------------------------
// Kernel B: rescaled linear-space pruning. One wave per block, 16 sites/wave.
// A operands register-double-buffered one step ahead; right-leaf B tiles are
// DMA'd into double-buffered LDS via the Tensor Data Mover and synchronized
// with s_wait_tensorcnt (in-order TENSORcnt => wait(1) releases current tile).
// ---------------------------------------------------------------------------
__global__ __launch_bounds__(32) void prune_kernel(
    const float* __restrict__ leaf, const float* __restrict__ stat_logits,
    const int* __restrict__ ops, const float* __restrict__ pswz,
    float* __restrict__ sitell)
{
    __shared__ float piS[20];
    __shared__ float V[32 * 16];                 // padded states x sites
    __shared__ float leafBuf[2][TILE_FLOATS];    // TDM staging, double buffered

    const int lane  = threadIdx.x;
    const int n     = lane & 15;
    const int kxoff = (lane < 16) ? 0 : 2;
    const int s0    = blockIdx.x * 16;

    if (lane == 0) {
        float mx = -1e30f;
        for (int i = 0; i < 20; ++i) mx = fmaxf(mx, stat_logits[i]);
        float sum = 0.f;
        for (int i = 0; i < 20; ++i) { float e = __expf(stat_logits[i] - mx); piS[i] = e; sum += e; }
        float inv = 1.f / sum;
        for (int i = 0; i < 20; ++i) piS[i] *= inv;
    }
    __syncthreads();

    // --- TDM: issue one 1.28KB leaf tile (16 sites x 20 states, contiguous) ---
    auto issueLeaf = [&](int leafNode, int parity) {
        unsigned long long ga = (unsigned long long)(uintptr_t)
            (leaf + ((size_t)leafNode * NUM_SITES + s0) * NUM_STATES);
        unsigned ldsA = (unsigned)(uintptr_t)(&leafBuf[parity][0]);
        v4u g0;
        g0.x = 1u;                                            // count=1 (valid user D#)
        g0.y = ldsA;                                          // lds_addr
        g0.z = (unsigned)(ga & 0xFFFFFFFFull);                // global_addr[31:0]
        g0.w = (unsigned)((ga >> 32) & 0x1FFFFFFull)          // global_addr[56:32]
             | (2u << 30);                                    // type=2 ("image")
        v8i g1;
        g1[0] = (int)(2u << 16);          // workgroup_mask=0, data_size=2 (4B)
        g1[1] = (int)(320u << 16);        // tensor_dim0[15:0] = 320
        g1[2] = (int)(1u << 16);          // tensor_dim0[31:16]=0, tensor_dim1[15:0]=1
        g1[3] = (int)(320u << 16);        // tensor_dim1[31:16]=0, tile_dim0=320
        g1[4] = 1;                        // tile_dim1=1, tile_dim2=0
        g1[5] = 320;                      // tensor_dim0_stride[31:0]
        g1[6] = 0;                        // stride0 hi / stride1 lo
        g1[7] = 0;                        // stride1 hi
        v4i gz4 = {0, 0, 0, 0};
        v8i gz8 = {0, 0, 0, 0, 0, 0, 0, 0};
        // amdgpu-toolchain (clang-23) 6-arg form: (g0, g1, g2, g3, gX, cpol)
        __builtin_amdgcn_tensor_load_to_lds(g0, g1, gz4, gz4, gz8, 0);
    };

    // A operands for one step: [0..4]=L tile0, [5..9]=L tile1, [10..14]=R t0, [15..19]=R t1
    v2f Acur[20], Anxt[20];
    auto loadA = [&](v2f* A, int nl_, int nr_) {
        const v2f* Al = (const v2f*)(pswz + (size_t)nl_ * PSWZ_PER_NODE);
        const v2f* Ar = (const v2f*)(pswz + (size_t)nr_ * PSWZ_PER_NODE);
#pragma unroll
        for (int c = 0; c < 5; ++c) {
            A[c]      = Al[c * 32 + lane];
            A[5 + c]  = Al[(5 + c) * 32 + lane];
            A[10 + c] = Ar[c * 32 + lane];
            A[15 + c] = Ar[(5 + c) * 32 + lane];
        }
    };

    // Prologue: stage step 0's right-leaf tile + step 0's A operands.
    {
        int nr0 = ops[2];
        issueLeaf(nr0, 0);
        loadA(Acur, ops[1], nr0);
    }

    float logscale = 0.f;
    v8f prod0 = {};
    v8f prod1 = {};

    for (int k = 0; k < NUM_OPS; ++k) {
        const int p = k & 1;

        // Pipeline next step: issue its TDM leaf transfer + A-operand loads,
        // then wait for the *current* tile (<=1 outstanding, in-order).
        if (k + 1 < NUM_OPS) {
            int nlN = ops[(k + 1) * 3 + 1];
            int nrN = ops[(k + 1) * 3 + 2];
            issueLeaf(nrN, p ^ 1);
            loadA(Anxt, nlN, nrN);
            __builtin_amdgcn_s_wait_tensorcnt((short)1);
        } else {
            __builtin_amdgcn_s_wait_tensorcnt((short)0);
        }

        // Left B operand: step 0 reads leaf 0 from global, others read V (LDS).
        v2f bl[5];
        if (k == 0) {
            const float* leafL = leaf + ((size_t)ops[1] * NUM_SITES + s0 + n) * NUM_STATES;
#pragma unroll
            for (int c = 0; c < 5; ++c)
                bl[c] = *(const v2f*)(leafL + 4 * c + kxoff);
        } else {
#pragma unroll
            for (int c = 0; c < 5; ++c) {
                const int kx = 4 * c + kxoff;
                bl[c].x = V[kx * 16 + n];
                bl[c].y = V[(kx + 1) * 16 + n];
            }
        }
        // Right B operand from the TDM-staged LDS tile (contiguous ds_load_b64).
        v2f br[5];
#pragma unroll
        for (int c = 0; c < 5; ++c)
            br[c] = *(const v2f*)(&leafBuf[p][n * NUM_STATES + 4 * c + kxoff]);

        v8f accL0 = {}, accL1 = {}, accR0 = {}, accR1 = {};
#pragma unroll
        for (int c = 0; c < 5; ++c) {
            accL0 = __builtin_amdgcn_wmma_f32_16x16x4_f32(false, Acur[c],      false, bl[c], (short)0, accL0, false, false);
            accL1 = __builtin_amdgcn_wmma_f32_16x16x4_f32(false, Acur[5 + c],  false, bl[c], (short)0, accL1, false, false);
            accR0 = __builtin_amdgcn_wmma_f32_16x16x4_f32(false, Acur[10 + c], false, br[c], (short)0, accR0, false, false);
            accR1 = __builtin_amdgcn_wmma_f32_16x16x4_f32(false, Acur[15 + c], false, br[c], (short)0, accR1, false, false);
        }

        prod0 = accL0 * accR0;
        prod1 = accL1 * accR1;   // rows >= 20 are zero (zero-padded P rows)

        // Per-site max rescale (column n lives in lanes n and n+16).
        float m = 0.f;
#pragma unroll
        for (int r = 0; r < 8; ++r) m = fmaxf(m, fmaxf(prod0[r], prod1[r]));
        m = fmaxf(m, __shfl_xor(m, 16, 32));
        if (m > 0.f) {
            const float inv = 1.f / m;
#pragma unroll
            for (int r = 0; r < 8; ++r) { prod0[r] *= inv; prod1[r] *= inv; }
            logscale += __logf(m);
        }

        __syncthreads();
        const int rb = (lane < 16) ? 0 : 8;
#pragma unroll
        for (int r = 0; r < 8; ++r) {
            V[(rb + r) * 16 + n]      = prod0[r];
            V[(16 + rb + r) * 16 + n] = prod1[r];
        }
        __syncthreads();

        if (k + 1 < NUM_OPS) {
#pragma unroll
            for (int i = 0; i < 20; ++i) Acur[i] = Anxt[i];
        }
    }

    // Root: ll = log( sum_i pi_i * V_i ) + logscale
    float part = 0.f;
    const int rb = (lane < 16) ? 0 : 8;
#pragma unroll
    for (int r = 0; r < 8; ++r) {
        part += piS[rb + r] * prod0[r];
        const int m1 = 16 + rb + r;
        if (m1 < 20) part += piS[m1] * prod1[r];
    }
    part += __shfl_xor(part, 16, 32);
    if (lane < 16) sitell[s0 + n] = __logf(part) + logscale;
}

// ---------------------------------------------------------------------------
// Kernel C: deterministic mean over 256 per-site log-likelihoods.
// ---------------------------------------------------------------------------
__global__ __launch_bounds__(256) void reduce_mean_kernel(
    const float* __restrict__ sitell, float* __restrict__ out)
{
    __shared__ float red[256];
    const int t = threadIdx.x;
    red[t] = sitell[t];
    __syncthreads();
    for (int s = 128; s > 0; s >>= 1) {
        if (t < s) red[t] += red[t + s];
        __syncthreads();
    }
    if (t == 0) out[0] = red[0] * (1.0f / (float)NUM_SITES);
}

extern "C" void kernel_launch(void* const* d_in, const int* in_sizes, int n_in,
                              void* d_out, int out_size, void* d_ws, size_t ws_size,
                              hipStream_t stream) {
    const float* leaf  = (const float*)d_in[0];   // (2048, 256, 20) f32
    const float* exch  = (const float*)d_in[1];   // (20, 20) f32
    const float* statl = (const float*)d_in[2];   // (20,) f32
    const float* blen  = (const float*)d_in[3];   // (4095,) f32
    const int*   ops   = (const int*)d_in[4];     // (2047, 3) i32
    float* out = (float*)d_out;

    float* pswz   = (float*)d_ws;                 // 4095 * 640 floats
    float* sitell = pswz + PSWZ_TOTAL;            // 256 floats

    hipLaunchKernelGGL(build_p_kernel, dim3(NUM_NODES), dim3(256), 0, stream,
                       exch, statl, blen, pswz);
    hipLaunchKernelGGL(prune_kernel, dim3(NUM_SITES / 16), dim3(32), 0, stream,
                       leaf, statl, ops, pswz, sitell);
    hipLaunchKernelGGL(reduce_mean_kernel, dim3(1), dim3(256), 0, stream,
                       sitell, out);
}